// CMAELoss_16277926052239
// MI455X (gfx1250) — compile-verified
//
#include <hip/hip_runtime.h>
#include <hip/hip_bf16.h>
#include <math.h>

typedef __attribute__((ext_vector_type(2))) float v2f;
typedef __attribute__((ext_vector_type(8))) float v8f;

#define N_CTR   128
#define D_CTR   256
#define TEMP    0.1f
#define COS_EPS 1e-8f

#define PATCH_P   768
#define N_PATCH   (128 * 196)       // 25088
#define WAVES_PB  8
#define RECON_BLOCKS ((N_PATCH + WAVES_PB - 1) / WAVES_PB)   // 3136

// workspace layout (float offsets)
#define WS_S      0                 // 128*128 raw similarity
#define WS_RINVP  16384             // 128 inverse norms of preds
#define WS_RINVT  16512             // 128 inverse norms of targets
#define WS_CONTR  16640             // contrastive scalar
#define WS_PART_L 16704             // recon loss partials (<=4096)
#define WS_PART_M (16704 + 4096)    // mask partials (<=4096)

// ---------------------------------------------------------------------------
// Kernel 1: inverse row norms of student/teacher. One wave per row (256 rows).
// ---------------------------------------------------------------------------
__global__ void norms_kernel(const float* __restrict__ sp,
                             const float* __restrict__ tp,
                             float* __restrict__ ws) {
    int wave = (blockIdx.x * blockDim.x + threadIdx.x) >> 5;   // 0..255
    int lane = threadIdx.x & 31;
    const float* src = (wave < N_CTR) ? (sp + wave * D_CTR)
                                      : (tp + (wave - N_CTR) * D_CTR);
    float ss = 0.f;
    for (int i = lane; i < D_CTR; i += 32) {
        float v = src[i];
        ss += v * v;
    }
    #pragma unroll
    for (int off = 16; off; off >>= 1) ss += __shfl_xor(ss, off, 32);
    if (lane == 0) {
        float nrm = fmaxf(sqrtf(ss), COS_EPS);
        float r = 1.0f / nrm;
        if (wave < N_CTR) ws[WS_RINVP + wave] = r;
        else              ws[WS_RINVT + (wave - N_CTR)] = r;
    }
}

// ---------------------------------------------------------------------------
// Kernel 2: raw S = preds @ targets^T via V_WMMA_F32_16X16X4_F32.
// One wave (32-thread block, EXEC all ones) per 16x16 tile; 8x8 tiles.
// A-frag layout (16x4 f32): lanes 0-15 hold K=0,1 of row M=lane;
// lanes 16-31 hold K=2,3. B mirrors with N=lane&15. C: VGPR r = row r (+8).
// ---------------------------------------------------------------------------
__global__ void gemm_kernel(const float* __restrict__ sp,
                            const float* __restrict__ tp,
                            float* __restrict__ ws) {
    int tile = blockIdx.x;          // 0..63
    int ti = tile >> 3;             // row tile
    int tj = tile & 7;              // col tile
    int lane = threadIdx.x;         // 0..31
    int half = lane >> 4;           // 0 or 1 -> K pair select
    int lid  = lane & 15;

    const float* arow = sp + (ti * 16 + lid) * D_CTR;  // pn row m
    const float* brow = tp + (tj * 16 + lid) * D_CTR;  // tn row n (B = tn^T)

    v8f c = {0.f, 0.f, 0.f, 0.f, 0.f, 0.f, 0.f, 0.f};
    for (int k = 0; k < D_CTR; k += 4) {
        int k0 = k + half * 2;
        v2f a, b;
        a.x = arow[k0]; a.y = arow[k0 + 1];
        b.x = brow[k0]; b.y = brow[k0 + 1];
        c = __builtin_amdgcn_wmma_f32_16x16x4_f32(
                /*neg_a=*/false, a, /*neg_b=*/false, b,
                /*c_mod=*/(short)0, c, /*reuse_a=*/false, /*reuse_b=*/false);
    }

    float* S = ws + WS_S;
    int col = tj * 16 + lid;
    #pragma unroll
    for (int r = 0; r < 8; ++r) {
        int row = ti * 16 + r + half * 8;
        S[row * N_CTR + col] = c[r];
    }
}

// ---------------------------------------------------------------------------
// Kernel 3: per-row scaled logsumexp + trace -> contrastive loss scalar.
// One block of 128 threads, thread i = row i. Scaling rinv_p[i]*rinv_t[j]/T
// fused here (rank-1 factor pulled out of the GEMM).
// ---------------------------------------------------------------------------
__global__ void lse_kernel(float* __restrict__ ws) {
    __shared__ float rt[N_CTR];
    __shared__ float red[N_CTR];
    __shared__ float redd[N_CTR];
    int i = threadIdx.x;            // 0..127
    rt[i] = ws[WS_RINVT + i];
    __syncthreads();

    const float* S = ws + WS_S + i * N_CTR;
    float rp = ws[WS_RINVP + i] * (1.0f / TEMP);

    float mx = -3.402823466e38f;
    for (int j = 0; j < N_CTR; ++j) {
        float s = S[j] * rp * rt[j];
        mx = fmaxf(mx, s);
    }
    float sum = 0.f, diag = 0.f;
    for (int j = 0; j < N_CTR; ++j) {
        float s = S[j] * rp * rt[j];
        sum += expf(s - mx);
        if (j == i) diag = s;
    }
    red[i]  = mx + logf(sum);
    redd[i] = diag;
    __syncthreads();
    #pragma unroll
    for (int s2 = 64; s2; s2 >>= 1) {
        if (i < s2) { red[i] += red[i + s2]; redd[i] += redd[i + s2]; }
        __syncthreads();
    }
    if (i == 0) ws[WS_CONTR] = (red[0] - redd[0]) / (float)N_CTR;
}

// ---------------------------------------------------------------------------
// Kernel 4: reconstruction loss, one wave per patch (768 floats = 6 float4
// per lane). Coalesced b128 loads, both patches kept in VGPRs, wave butterfly
// reductions, deterministic per-block partials (no float atomics).
// ---------------------------------------------------------------------------
__global__ void recon_kernel(const float* __restrict__ tgt,
                             const float* __restrict__ prd,
                             const float* __restrict__ mask,
                             float* __restrict__ ws) {
    int wid  = threadIdx.x >> 5;
    int lane = threadIdx.x & 31;
    int patch = blockIdx.x * WAVES_PB + wid;

    float loss = 0.f, mval = 0.f;
    if (patch < N_PATCH) {
        const float4* t4 = (const float4*)(tgt + (size_t)patch * PATCH_P);
        const float4* p4 = (const float4*)(prd + (size_t)patch * PATCH_P);
        float4 tv[6], pv[6];
        #pragma unroll
        for (int i = 0; i < 6; ++i) {
            tv[i] = t4[i * 32 + lane];
            pv[i] = p4[i * 32 + lane];
        }
        float s = 0.f, ss = 0.f;
        #pragma unroll
        for (int i = 0; i < 6; ++i) {
            s  += tv[i].x + tv[i].y + tv[i].z + tv[i].w;
            ss += tv[i].x * tv[i].x + tv[i].y * tv[i].y
                + tv[i].z * tv[i].z + tv[i].w * tv[i].w;
        }
        #pragma unroll
        for (int off = 16; off; off >>= 1) {
            s  += __shfl_xor(s,  off, 32);
            ss += __shfl_xor(ss, off, 32);
        }
        float mean = s * (1.0f / (float)PATCH_P);
        float var  = (ss - (float)PATCH_P * mean * mean) * (1.0f / (float)(PATCH_P - 1));
        float rstd = rsqrtf(var + 1e-6f);

        float d2 = 0.f;
        #pragma unroll
        for (int i = 0; i < 6; ++i) {
            float dx = pv[i].x - (tv[i].x - mean) * rstd; d2 += dx * dx;
            float dy = pv[i].y - (tv[i].y - mean) * rstd; d2 += dy * dy;
            float dz = pv[i].z - (tv[i].z - mean) * rstd; d2 += dz * dz;
            float dw = pv[i].w - (tv[i].w - mean) * rstd; d2 += dw * dw;
        }
        #pragma unroll
        for (int off = 16; off; off >>= 1) d2 += __shfl_xor(d2, off, 32);

        float m = mask[patch];
        loss = d2 * (1.0f / (float)PATCH_P) * m;
        mval = m;
    }

    __shared__ float lred[WAVES_PB], mred[WAVES_PB];
    if (lane == 0) { lred[wid] = loss; mred[wid] = mval; }
    __syncthreads();
    if (threadIdx.x == 0) {
        float L = 0.f, M = 0.f;
        #pragma unroll
        for (int w = 0; w < WAVES_PB; ++w) { L += lred[w]; M += mred[w]; }
        ws[WS_PART_L + blockIdx.x] = L;
        ws[WS_PART_M + blockIdx.x] = M;
    }
}

// ---------------------------------------------------------------------------
// Kernel 5: reduce partials, combine with contrastive scalar, emit 3 outputs.
// ---------------------------------------------------------------------------
__global__ void final_kernel(const float* __restrict__ ws,
                             float* __restrict__ out) {
    __shared__ float lred[256], mred[256];
    int t = threadIdx.x;
    float L = 0.f, M = 0.f;
    for (int i = t; i < RECON_BLOCKS; i += 256) {
        L += ws[WS_PART_L + i];
        M += ws[WS_PART_M + i];
    }
    lred[t] = L; mred[t] = M;
    __syncthreads();
    #pragma unroll
    for (int s = 128; s; s >>= 1) {
        if (t < s) { lred[t] += lred[t + s]; mred[t] += mred[t + s]; }
        __syncthreads();
    }
    if (t == 0) {
        float recon = lred[0] / mred[0];
        float contr = ws[WS_CONTR];
        out[0] = recon;
        out[1] = contr;
        out[2] = recon + contr;
    }
}

extern "C" void kernel_launch(void* const* d_in, const int* in_sizes, int n_in,
                              void* d_out, int out_size, void* d_ws, size_t ws_size,
                              hipStream_t stream) {
    const float* student = (const float*)d_in[0];   // [128, 256]
    const float* teacher = (const float*)d_in[1];   // [128, 256]
    const float* tgt     = (const float*)d_in[2];   // [128, 196, 768]
    const float* prd     = (const float*)d_in[3];   // [128, 196, 768]
    const float* mask    = (const float*)d_in[4];   // [128, 196]
    float* out = (float*)d_out;
    float* ws  = (float*)d_ws;
    (void)in_sizes; (void)n_in; (void)out_size; (void)ws_size;

    // 256 row-norm waves -> 32 blocks x 256 threads
    norms_kernel<<<32, 256, 0, stream>>>(student, teacher, ws);
    // 64 tiles, one wave each (EXEC all ones for WMMA)
    gemm_kernel<<<64, 32, 0, stream>>>(student, teacher, ws);
    // single block, one thread per row
    lse_kernel<<<1, 128, 0, stream>>>(ws);
    // bandwidth-dominant streaming reduction
    recon_kernel<<<RECON_BLOCKS, 256, 0, stream>>>(tgt, prd, mask, ws);
    final_kernel<<<1, 256, 0, stream>>>(ws, out);
}